// OptimizedQuaternionSSM_65352222376533
// MI455X (gfx1250) — compile-verified
//
#include <hip/hip_runtime.h>
#include <math.h>

// Problem constants (match reference)
#define BSZ   2
#define LEN   2048
#define DQ    512
#define SQ    4
#define NCH   16                 // chunks per sequence
#define CT    (LEN / NCH)        // 128 steps per chunk
#define EPSQ  1e-6f
#define CHAINS (BSZ * DQ * SQ)   // 4096 independent recurrences

typedef float v2f __attribute__((ext_vector_type(2)));
typedef float v8f __attribute__((ext_vector_type(8)));
typedef int   v4i __attribute__((ext_vector_type(4)));

#define AS1 __attribute__((address_space(1)))
#define AS3 __attribute__((address_space(3)))

struct Quat { float r, i, j, k; };

__device__ __forceinline__ Quat qmul(const Quat a, const Quat b) {
    Quat o;
    o.r = a.r*b.r - a.i*b.i - a.j*b.j - a.k*b.k;
    o.i = a.r*b.i + a.i*b.r + a.j*b.k - a.k*b.j;
    o.j = a.r*b.j - a.i*b.k + a.j*b.r + a.k*b.i;
    o.k = a.r*b.k + a.i*b.j - a.j*b.i + a.k*b.r;
    return o;
}

__device__ __forceinline__ Quat ld4(const float* p) {
    float4 v = *(const float4*)p;
    return Quat{v.x, v.y, v.z, v.w};
}

// Async global->LDS copy of 16 bytes (ASYNCcnt-tracked on gfx1250).
__device__ __forceinline__ void async_copy16(float4* lds_dst, const float4* gsrc) {
#if __has_builtin(__builtin_amdgcn_global_load_async_to_lds_b128)
    __builtin_amdgcn_global_load_async_to_lds_b128(
        (AS1 v4i*)gsrc, (AS3 v4i*)lds_dst, 0, 0);
#else
    unsigned loff = (unsigned)(uintptr_t)(AS3 void*)lds_dst;
    asm volatile("global_load_async_to_lds_b128 %0, %1, off"
                 :: "v"(loff), "v"(gsrc) : "memory");
#endif
}

__device__ __forceinline__ void async_wait0() {
#if __has_builtin(__builtin_amdgcn_s_wait_asynccnt)
    __builtin_amdgcn_s_wait_asynccnt(0);
#else
    asm volatile("s_wait_asynccnt 0" ::: "memory");
#endif
}

// Branch-free 4-way component select (3 v_cndmask).
__device__ __forceinline__ float sel4(float4 c, bool b1, bool b2) {
    float lo = b1 ? c.y : c.x;
    float hi = b1 ? c.w : c.z;
    return b2 ? hi : lo;
}

// Per-timestep Cayley transform: builds transition quaternion q and input Bu.
__device__ __forceinline__ void cayley_step(float dtv, Quat A, Quat Bq, Quat uq,
                                            Quat& q, Quat& Bu) {
    Quat z  {dtv * A.r, dtv * A.i, dtv * A.j, dtv * A.k};
    Quat num{1.0f + 0.5f * z.r,  0.5f * z.i,  0.5f * z.j,  0.5f * z.k};
    Quat den{1.0f - 0.5f * z.r, -0.5f * z.i, -0.5f * z.j, -0.5f * z.k};
    float n2 = den.r*den.r + den.i*den.i + den.j*den.j + den.k*den.k + EPSQ;
    float s  = 1.0f / n2;
    Quat dinv{den.r * s, -den.i * s, -den.j * s, -den.k * s};
    q = qmul(dinv, num);
    Quat dinv_dt{dinv.r * dtv, dinv.i * dtv, dinv.j * dtv, dinv.k * dtv};
    Quat dB = qmul(dinv_dt, Bq);
    Bu = qmul(dB, uq);
}

// ---------------------------------------------------------------------------
// Phase 1: per-chunk summaries.  One thread per (b, chunk, d); carries all 4 s.
// B_in rows for the chunk are staged block-wide into LDS via async copies.
// ---------------------------------------------------------------------------
__global__ __launch_bounds__(256)
void qssm_phase1(const float* __restrict__ u, const float* __restrict__ dtg,
                 const float* __restrict__ Bin,
                 const float* __restrict__ Alog, const float* __restrict__ Ai,
                 const float* __restrict__ Aj,   const float* __restrict__ Ak,
                 float* __restrict__ Qbuf, float* __restrict__ Vbuf)
{
    const int gid = blockIdx.x * blockDim.x + threadIdx.x;
    const int d = gid % DQ;
    const int c = (gid / DQ) % NCH;        // block-uniform
    const int b = gid / (DQ * NCH);        // block-uniform
    const int t0 = c * CT;

    const float4* u4 = (const float4*)u;
    const float4* B4 = (const float4*)Bin;

    // Stage this chunk's B_in [CT x SQ] quaternions into LDS (8 KB).
    __shared__ float4 sB[CT * SQ];
    {
        const float4* gB = B4 + (b * LEN + t0) * SQ;
        for (int e = threadIdx.x; e < CT * SQ; e += 256)
            async_copy16(&sB[e], gB + e);
        async_wait0();
        __syncthreads();
    }

    Quat A[SQ], h[SQ], Q[SQ];
    #pragma unroll
    for (int s = 0; s < SQ; ++s) {
        A[s] = Quat{-expf(Alog[d*SQ + s]), Ai[d*SQ + s], Aj[d*SQ + s], Ak[d*SQ + s]};
        h[s] = Quat{0.f, 0.f, 0.f, 0.f};
        Q[s] = Quat{1.f, 0.f, 0.f, 0.f};
    }

    for (int tt = 0; tt < CT; ++tt) {
        const int t   = t0 + tt;
        const int row = (b * LEN + t) * DQ;
        const float dtv = dtg[row + d];
        float4 uv = u4[row + d];
        Quat uq{uv.x, uv.y, uv.z, uv.w};
        if (tt + 8 < CT)
            __builtin_prefetch((const void*)(u4 + row + 8*DQ + d), 0, 1);
        #pragma unroll
        for (int s = 0; s < SQ; ++s) {
            float4 bv = sB[tt * SQ + s];
            Quat Bq{bv.x, bv.y, bv.z, bv.w};
            Quat q, Bu;
            cayley_step(dtv, A[s], Bq, uq, q, Bu);
            Quat nh = qmul(q, h[s]);
            h[s] = Quat{nh.r + Bu.r, nh.i + Bu.i, nh.j + Bu.j, nh.k + Bu.k};
            Q[s] = qmul(q, Q[s]);
        }
    }

    #pragma unroll
    for (int s = 0; s < SQ; ++s) {
        const int chain = (b * DQ + d) * SQ + s;
        const int idx   = chain * NCH + c;
        ((float4*)Qbuf)[idx] = make_float4(Q[s].r, Q[s].i, Q[s].j, Q[s].k);
        ((float4*)Vbuf)[idx] = make_float4(h[s].r, h[s].i, h[s].j, h[s].k);
    }
}

// ---------------------------------------------------------------------------
// Phase 2: tiny inter-chunk exclusive scan per chain (16 steps, 4096 threads).
// ---------------------------------------------------------------------------
__global__ __launch_bounds__(256)
void qssm_phase2(const float* __restrict__ Qbuf, const float* __restrict__ Vbuf,
                 float* __restrict__ Hstart)
{
    const int chain = blockIdx.x * blockDim.x + threadIdx.x;
    Quat h{0.f, 0.f, 0.f, 0.f};
    for (int c = 0; c < NCH; ++c) {
        const int idx = chain * NCH + c;
        ((float4*)Hstart)[idx] = make_float4(h.r, h.i, h.j, h.k);
        Quat Q = ld4(Qbuf + 4*idx);
        Quat V = ld4(Vbuf + 4*idx);
        Quat qh = qmul(Q, h);
        h = Quat{qh.r + V.r, qh.i + V.i, qh.j + V.j, qh.k + V.k};
    }
}

// ---------------------------------------------------------------------------
// Phase 3: finalize.  One wave handles (b, chunk, 32 d-channels); lane owns d.
// y = sum_s L(C_s) @ h via V_WMMA_F32_16X16X4_F32.  A-operand selection is
// fully branch-free (loop-invariant per-lane index/sign masks), so EXEC stays
// all-ones through the WMMAs.  B_in/C_in chunk rows staged via async LDS.
// ---------------------------------------------------------------------------
__global__ __launch_bounds__(256)
void qssm_phase3(const float* __restrict__ u, const float* __restrict__ dtg,
                 const float* __restrict__ Bin, const float* __restrict__ Cin,
                 const float* __restrict__ Alog, const float* __restrict__ Ai,
                 const float* __restrict__ Aj,   const float* __restrict__ Ak,
                 const float* __restrict__ Hstart, float* __restrict__ out)
{
    const int tid  = blockIdx.x * blockDim.x + threadIdx.x;
    const int lane = threadIdx.x & 31;
    const int wave = tid >> 5;
    const int ndt  = DQ / 32;                       // 16 d-tiles
    const int dtile = wave % ndt;
    const int c     = (wave / ndt) % NCH;           // block-uniform
    const int b     = wave / (ndt * NCH);           // block-uniform
    const int dbase = dtile * 32;
    const int d     = dbase + lane;
    const int t0    = c * CT;

    const float4* u4 = (const float4*)u;
    const float4* B4 = (const float4*)Bin;
    const float4* C4 = (const float4*)Cin;
    float4* out4 = (float4*)out;

    // Stage this chunk's B_in and C_in rows into LDS (16 KB) asynchronously.
    __shared__ float4 sB[CT * SQ];
    __shared__ float4 sC[CT * SQ];
    {
        const float4* gB = B4 + (b * LEN + t0) * SQ;
        const float4* gC = C4 + (b * LEN + t0) * SQ;
        for (int e = threadIdx.x; e < CT * SQ; e += 256) {
            async_copy16(&sB[e], gB + e);
            async_copy16(&sC[e], gC + e);
        }
        async_wait0();
        __syncthreads();
    }

    Quat A[SQ], h[SQ];
    #pragma unroll
    for (int s = 0; s < SQ; ++s) {
        A[s] = Quat{-expf(Alog[d*SQ + s]), Ai[d*SQ + s], Aj[d*SQ + s], Ak[d*SQ + s]};
        const int chain = (b * DQ + d) * SQ + s;
        h[s] = ld4(Hstart + 4 * (chain * NCH + c));
    }

    const int  m     = lane & 15;       // WMMA row (output quat component if <4)
    const bool hih   = lane >= 16;      // upper lane half -> K = 2,3
    const int  srcLo = m;               // owner lane of d = dbase + m
    const int  srcHi = m + 16;          // owner lane of d = dbase + 16 + m

    // Loop-invariant per-lane A-operand selectors: which C component feeds
    // ax/ay, its sign, and whether this row (m<4) is a valid L(C) row.
    const int mm = m & 3;
    int ix, iy; unsigned sxm, sym;
    if (!hih) { ix = mm;     iy = mm ^ 1;
                sxm = 0u;
                sym = (mm == 0 || mm == 3) ? 0x80000000u : 0u; }
    else      { ix = mm ^ 2; iy = mm ^ 3;
                sxm = (mm < 2) ? 0x80000000u : 0u;
                sym = (mm == 0 || mm == 2) ? 0x80000000u : 0u; }
    const bool ix1 = ix & 1, ix2 = (ix & 2) != 0;
    const bool iy1 = iy & 1, iy2 = (iy & 2) != 0;
    const float amask = (m < 4) ? 1.0f : 0.0f;

    for (int tt = 0; tt < CT; ++tt) {
        const int t   = t0 + tt;
        const int row = (b * LEN + t) * DQ;
        const float dtv = dtg[row + d];
        float4 uv = u4[row + d];
        Quat uq{uv.x, uv.y, uv.z, uv.w};
        if (tt + 8 < CT)
            __builtin_prefetch((const void*)(u4 + row + 8*DQ + d), 0, 1);

        v8f acc_lo = {0.f,0.f,0.f,0.f,0.f,0.f,0.f,0.f};
        v8f acc_hi = {0.f,0.f,0.f,0.f,0.f,0.f,0.f,0.f};

        #pragma unroll
        for (int s = 0; s < SQ; ++s) {
            float4 bv = sB[tt * SQ + s];
            Quat Bq{bv.x, bv.y, bv.z, bv.w};
            Quat q, Bu;
            cayley_step(dtv, A[s], Bq, uq, q, Bu);
            Quat nh = qmul(q, h[s]);
            h[s] = Quat{nh.r + Bu.r, nh.i + Bu.i, nh.j + Bu.j, nh.k + Bu.k};

            // --- A operand: L(C_s), 16x4 f32 layout, branch-free build.
            float4 cv = sC[tt * SQ + s];
            float ax = amask * __uint_as_float(
                           __float_as_uint(sel4(cv, ix1, ix2)) ^ sxm);
            float ay = amask * __uint_as_float(
                           __float_as_uint(sel4(cv, iy1, iy2)) ^ sym);
            v2f amat = {ax, ay};

            // --- B operand (4x16 f32): column n = d-channel, row k = h comp.
            float l0 = __shfl(h[s].r, srcLo, 32);
            float l1 = __shfl(h[s].i, srcLo, 32);
            float l2 = __shfl(h[s].j, srcLo, 32);
            float l3 = __shfl(h[s].k, srcLo, 32);
            v2f blo = { hih ? l2 : l0, hih ? l3 : l1 };
            float g0 = __shfl(h[s].r, srcHi, 32);
            float g1 = __shfl(h[s].i, srcHi, 32);
            float g2 = __shfl(h[s].j, srcHi, 32);
            float g3 = __shfl(h[s].k, srcHi, 32);
            v2f bhi = { hih ? g2 : g0, hih ? g3 : g1 };

            acc_lo = __builtin_amdgcn_wmma_f32_16x16x4_f32(
                         false, amat, false, blo, (short)0, acc_lo, false, false);
            acc_hi = __builtin_amdgcn_wmma_f32_16x16x4_f32(
                         false, amat, false, bhi, (short)0, acc_hi, false, false);
        }

        // D layout: lane n (0..15) holds M=0..3 (y comps) in VGPRs 0..3.
        if (lane < 16) {
            out4[row + dbase + lane]      = make_float4(acc_lo[0], acc_lo[1], acc_lo[2], acc_lo[3]);
            out4[row + dbase + 16 + lane] = make_float4(acc_hi[0], acc_hi[1], acc_hi[2], acc_hi[3]);
        }
    }
}

// ---------------------------------------------------------------------------
extern "C" void kernel_launch(void* const* d_in, const int* in_sizes, int n_in,
                              void* d_out, int out_size, void* d_ws, size_t ws_size,
                              hipStream_t stream)
{
    const float* u    = (const float*)d_in[0];
    const float* dtg  = (const float*)d_in[1];
    const float* Bin  = (const float*)d_in[2];
    const float* Cin  = (const float*)d_in[3];
    const float* Alog = (const float*)d_in[4];
    const float* Ai   = (const float*)d_in[5];
    const float* Aj   = (const float*)d_in[6];
    const float* Ak   = (const float*)d_in[7];
    float* out = (float*)d_out;

    // Workspace: 3 buffers of CHAINS*NCH quaternions (1 MB each)
    float* qb = (float*)d_ws;
    float* vb = qb + (size_t)CHAINS * NCH * 4;
    float* hs = vb + (size_t)CHAINS * NCH * 4;

    qssm_phase1<<<(BSZ * NCH * DQ) / 256, 256, 0, stream>>>(
        u, dtg, Bin, Alog, Ai, Aj, Ak, qb, vb);
    qssm_phase2<<<CHAINS / 256, 256, 0, stream>>>(qb, vb, hs);
    qssm_phase3<<<(BSZ * NCH * (DQ / 32) * 32) / 256, 256, 0, stream>>>(
        u, dtg, Bin, Cin, Alog, Ai, Aj, Ak, hs, out);
}